// PatchWiseConv2d_83854941487370
// MI455X (gfx1250) — compile-verified
//
#include <hip/hip_runtime.h>

// CDNA5 (gfx1250) wave32 WMMA types
typedef __attribute__((ext_vector_type(2))) float v2f;
typedef __attribute__((ext_vector_type(8))) float v8f;
typedef __attribute__((ext_vector_type(2))) int   v2i;

namespace {
constexpr int KH = 3, KW = 3;
constexpr int OH = 62, OW = 62;
constexpr int CI = 32;                  // input channels
constexpr int CO = 64;                  // output channels
constexpr int IH = 64, IW = 64;
constexpr int IHW   = IH * IW;          // 4096  x channel stride
constexpr int OHW   = OH * OW;          // 3844  weight/out spatial stride
constexpr int WKSTR = KH * KW * OHW;    // 34596 weight input-channel stride

constexpr int TW   = 16;                // w-positions per block (one per wave)
constexpr int KC   = 8;                 // k-chunk (channels per stage)
constexpr int NCHK = (CI * KH * KW) / KC; // 36 chunks
constexpr int ROWW  = 18;               // padded LDS w-row (floats): 8B aligned
constexpr int KKSTR = 64 * ROWW + 16;   // 1168: 2*KKSTR % 64 == 32 -> halves never collide
constexpr int BUFSZ = KC * KKSTR;       // 9344 floats per buffer
constexpr int NTHREADS = TW * 32;       // 512 = 16 waves
}

// ---- async global->LDS copy (CDNA5), with portable fallback --------------
// Builtin signature (per hipcc diagnostic): param0 = int2 __device__* (global),
// so pass generic non-const int2* pointers and let clang do the AS conversion.
__device__ __forceinline__ void async_copy_b64(const float* g, float* l) {
#if __has_builtin(__builtin_amdgcn_global_load_async_to_lds_b64)
  __builtin_amdgcn_global_load_async_to_lds_b64((v2i*)g, (v2i*)l, 0, 0);
#else
  *(v2f*)l = *(const v2f*)g;            // global_load_b64 + ds_store_b64
#endif
}

__device__ __forceinline__ void wait_async_all() {
#if __has_builtin(__builtin_amdgcn_s_wait_asynccnt)
  __builtin_amdgcn_s_wait_asynccnt(0);
#elif __has_builtin(__builtin_amdgcn_global_load_async_to_lds_b64)
  asm volatile("s_wait_asynccnt 0x0" ::: "memory");
#endif
}

// One wave == one (h,w): GEMM [16b x 288k] x [288k x 64o] -> [16 x 64] via
// V_WMMA_F32_16X16X4_F32; weights staged coalesced (along w) into LDS.
__global__ __launch_bounds__(NTHREADS)
void patchconv_wmma_lds(const float* __restrict__ x,
                        const float* __restrict__ wgt,
                        const float* __restrict__ bias,
                        float* __restrict__ out)
{
    __shared__ float sB[2 * BUFSZ];

    const int h   = blockIdx.x;
    const int w0  = min((int)blockIdx.y * TW, OW - TW);  // overlap last tile
    const int tid = threadIdx.x;
    const int wv  = tid >> 5;            // wave id == local w
    const int w   = w0 + wv;             // absolute w, always < OW
    const int lane = tid & 31;
    const int col  = lane & 15;          // A: batch m | B/D: column n
    const int half = lane >> 4;          // A/B: k-pair | C/D: m-block

    // Per-lane A base (x gathers; x is small + heavily L1/L2 reused)
    const float* pA = x + (size_t)col * (CI * IHW)
                        + (size_t)(2 * half) * IHW
                        + (size_t)h * IW + w;
    // Per-lane LDS read base for B (buffer 0)
    const float* sBl = sB + (size_t)(2 * half) * KKSTR
                          + (size_t)col * ROWW + wv;
    // Block-uniform weight base at (h, w0)
    const float* wb = wgt + (size_t)h * OW + w0;

    // Staging: per chunk move 8k x 64o x 16w; thread -> 8 b64 copies, o += 8 apart
    const int s_w2  = (tid & 7) * 2;     // even -> 8B-aligned globally & in LDS
    const int s_row = tid >> 3;          // 64 rows per pass, 8 passes
    const int s_o   = s_row >> 3;
    const int s_kk  = s_row & 7;

    auto stage = [&](int c, float* dst) {
        const int ko = c >> 2, i0 = (c & 3) * 8;
        const float* g0 = wb + (size_t)ko * OHW + (size_t)(i0 + s_kk) * WKSTR
                             + (size_t)s_o * (CI * WKSTR) + s_w2;
        float* l0 = dst + s_kk * KKSTR + s_o * ROWW + s_w2;
#pragma unroll
        for (int p = 0; p < 8; ++p) {    // row += 64 => o += 8 each pass
            async_copy_b64(g0 + (size_t)p * 8 * (CI * WKSTR), l0 + p * 8 * ROWW);
        }
    };

    v8f acc[4];
#pragma unroll
    for (int q = 0; q < 4; ++q) acc[q] = (v8f)0.0f;

    stage(0, sB);
    wait_async_all();
    __syncthreads();

#pragma unroll
    for (int c = 0; c < NCHK; ++c) {
        if (c + 1 < NCHK) stage(c + 1, sB + ((c + 1) & 1) * BUFSZ);

        const int ko = c >> 2, i0 = (c & 3) * 8;
        const int ky = ko / 3, kx = ko % 3;
        const float* sBc = sBl + (c & 1) * BUFSZ;
#pragma unroll
        for (int s = 0; s < 2; ++s) {
            const int ch0 = i0 + 4 * s;
            v2f a;                       // A 16x4: lanes0-15 k{0,1}, lanes16-31 k{2,3}
            a.x = pA[(ch0 + 0) * IHW + ky * IW + kx];
            a.y = pA[(ch0 + 1) * IHW + ky * IW + kx];
#pragma unroll
            for (int q = 0; q < 4; ++q) {
                v2f b;                   // bank-conflict-free ds_load_b32 pair
                b.x = sBc[(4 * s) * KKSTR + (q * 16) * ROWW];
                b.y = sBc[(4 * s + 1) * KKSTR + (q * 16) * ROWW];
                acc[q] = __builtin_amdgcn_wmma_f32_16x16x4_f32(
                    false, a, false, b, (short)0, acc[q], false, false);
            }
        }
        wait_async_all();   // chunk c+1 landed; buffers safe to swap
        __syncthreads();
    }

    // Epilogue: D vgpr r -> m = r + 8*half, lane col -> n; bias uniform over m.
    const float* pBias = bias + (size_t)col * OHW + (size_t)h * OW + w;
    float*       pO    = out  + (size_t)(8 * half) * (CO * OHW)
                              + (size_t)col * OHW
                              + (size_t)h * OW + w;
#pragma unroll
    for (int q = 0; q < 4; ++q) {
        const float bv = pBias[q * 16 * OHW];
#pragma unroll
        for (int r = 0; r < 8; ++r) {
            pO[(size_t)r * (CO * OHW) + (size_t)(q * 16) * OHW] = acc[q][r] + bv;
        }
    }
}

extern "C" void kernel_launch(void* const* d_in, const int* in_sizes, int n_in,
                              void* d_out, int out_size, void* d_ws, size_t ws_size,
                              hipStream_t stream) {
    const float* x    = (const float*)d_in[0];  // (16, 32, 64, 64)
    const float* wgt  = (const float*)d_in[1];  // (64, 32, 3, 3, 62, 62)
    const float* bias = (const float*)d_in[2];  // (1, 64, 62, 62)
    float* out = (float*)d_out;                 // (16, 64, 62, 62)

    dim3 grid(OH, (OW + TW - 1) / TW, 1);       // (62, 4)
    dim3 block(NTHREADS, 1, 1);                 // 512 = 16 waves
    patchconv_wmma_lds<<<grid, block, 0, stream>>>(x, wgt, bias, out);
}